// NLP_50216757625552
// MI455X (gfx1250) — compile-verified
//
#include <hip/hip_runtime.h>
#include <hip/hip_bf16.h>

// ---------------------------------------------------------------------------
// Normalized Laplacian Pyramid (NLP) for MI455X / gfx1250.
//  - fp32 throughout (matches reference; problem is memory-bound, not math-bound)
//  - separable 5-tap stride-2 downsample done with V_WMMA_F32_16X16X4_F32:
//      Down(16x16) = Mv(16x36) * X(36x48) * Mh^T(36x16), banded constant M in A
//  - up-sample + z + 5x5 divisive pool + normalize fused in one LDS-tiled kernel
//  - gamma x^(1/2.6) = v_exp_f32(g * v_log_f32(x)) via amdgcn builtins (x > 0)
// ---------------------------------------------------------------------------

typedef float v2f __attribute__((ext_vector_type(2)));
typedef float v8f __attribute__((ext_vector_type(8)));

#define GAMMA_EXP 0.38461538461538464f   // 1/2.6
#define SIGMA_P   0.17f
#define SIGMA_LP  4.86f

// reflect-101 index (jnp.pad mode='reflect'), clamped for the padded/unused taps
__device__ __forceinline__ int reflc(int r, int n) {
    r = (r < 0) ? -r : r;
    if (r >= n) r = 2 * n - 2 - r;
    if (r < 0)  r = 0;
    if (r >= n) r = n - 1;
    return r;
}

// 5-tap Laplacian kernel L = {0.05, 0.25, 0.4, 0.25, 0.05}, zero outside
__device__ __forceinline__ float ltap(int d) {
    float v = 0.0f;
    if (d == 0 || d == 4)      v = 0.05f;
    else if (d == 1 || d == 3) v = 0.25f;
    else if (d == 2)           v = 0.40f;
    return v;
}

// Source fetch for level l: l==0 -> NHWC image + gamma; l>0 -> previous down
// buffer (dims (H+2)x(W+2)) read with a (+1,+1) crop offset.
__device__ __forceinline__ float src_at(const float* __restrict__ src, int p,
                                        int y, int x, int H, int W, int gam) {
    if (gam) {
        int b = p / 3, c = p - 3 * b;
        float v = src[(((size_t)b * H + y) * W + x) * 3 + c];
        // v > 0 always (luminance in [5,180]) -> v_log_f32/v_exp_f32 HW path
        return __builtin_amdgcn_exp2f(GAMMA_EXP * __builtin_amdgcn_logf(v));
    } else {
        int Wp = W + 2;
        return src[(size_t)p * (H + 2) * Wp + (size_t)(y + 1) * Wp + (x + 1)];
    }
}

// ---------------------------------------------------------------------------
// Downsample: down[i][j] = sum_{k1,k2} L[k1] L[k2] x[refl(2i+k1-4)][refl(2j+k2-4)]
// One wave computes one 16x16 tile of the (H/2+2)x(W/2+2) down plane via WMMA.
// Pass 1: V(16x48) = Mv(16x36) * X(36x48)  (3 N-subtiles, 9 K-chunks each)
// Pass 2: Down^T(16x16) = Mh(16x36) * V^T(36x16)  (V staged through LDS)
// ---------------------------------------------------------------------------
__global__ void __launch_bounds__(32)
nlp_down(const float* __restrict__ src, float* __restrict__ down,
         int H, int W, int gammaFlag) {
    __shared__ float vbuf[16][49];   // 49 stride: avoid LDS bank conflicts

    const int H2 = H / 2 + 2, W2 = W / 2 + 2;
    const int p  = blockIdx.z;
    const int i0 = blockIdx.y * 16, j0 = blockIdx.x * 16;
    const int lane = threadIdx.x;
    const int n = lane & 15;          // A: row M / B: col N / D: col N
    const int half = lane >> 4;       // selects K pair (A/B) or M+8 (D)

    // reflected global columns for the 3 B-subtiles (fixed per lane)
    int colx[3];
#pragma unroll
    for (int nt = 0; nt < 3; ++nt)
        colx[nt] = reflc(2 * j0 - 4 + nt * 16 + n, W);

    v8f acc0 = {0,0,0,0,0,0,0,0};
    v8f acc1 = {0,0,0,0,0,0,0,0};
    v8f acc2 = {0,0,0,0,0,0,0,0};

    // ---- pass 1: vertical 5-tap stride-2 (contract over 36 slab rows) ----
    for (int c = 0; c < 9; ++c) {
        const int kb = 4 * c + 2 * half;              // K index of a.x / b.x
        v2f a;                                        // banded Mv[m][s]=L[s-2m]
        a.x = ltap(kb - 2 * n);
        a.y = ltap(kb + 1 - 2 * n);
        const int r0 = reflc(2 * i0 + kb - 4, H);
        const int r1 = reflc(2 * i0 + kb - 3, H);

        v2f b0, b1, b2;
        b0.x = src_at(src, p, r0, colx[0], H, W, gammaFlag);
        b0.y = src_at(src, p, r1, colx[0], H, W, gammaFlag);
        b1.x = src_at(src, p, r0, colx[1], H, W, gammaFlag);
        b1.y = src_at(src, p, r1, colx[1], H, W, gammaFlag);
        b2.x = src_at(src, p, r0, colx[2], H, W, gammaFlag);
        b2.y = src_at(src, p, r1, colx[2], H, W, gammaFlag);

        acc0 = __builtin_amdgcn_wmma_f32_16x16x4_f32(false, a, false, b0,
                                                     (short)0, acc0, false, false);
        acc1 = __builtin_amdgcn_wmma_f32_16x16x4_f32(false, a, false, b1,
                                                     (short)0, acc1, false, false);
        acc2 = __builtin_amdgcn_wmma_f32_16x16x4_f32(false, a, false, b2,
                                                     (short)0, acc2, false, false);
    }

    // stage V in LDS: D layout -> VGPR v is row (v + 8*half), lane is col n
#pragma unroll
    for (int v = 0; v < 8; ++v) {
        vbuf[v + 8 * half][n]      = acc0[v];
        vbuf[v + 8 * half][16 + n] = acc1[v];
        vbuf[v + 8 * half][32 + n] = acc2[v];
    }
    __syncthreads();

    // ---- pass 2: horizontal 5-tap stride-2 (contract over 36 slab cols) ----
    v8f dacc = {0,0,0,0,0,0,0,0};
    for (int c = 0; c < 9; ++c) {
        const int kb = 4 * c + 2 * half;
        v2f a;                                        // banded Mh[j][t]=L[t-2j]
        a.x = ltap(kb - 2 * n);
        a.y = ltap(kb + 1 - 2 * n);
        v2f b;                                        // B[t][m] = V[m][t]
        b.x = vbuf[n][kb];
        b.y = vbuf[n][kb + 1];
        dacc = __builtin_amdgcn_wmma_f32_16x16x4_f32(false, a, false, b,
                                                     (short)0, dacc, false, false);
    }

    // D layout: lane n = down-tile row m, VGPR v -> down-tile col j = v+8*half
    const int gi = i0 + n;
    if (gi < H2) {
        float* dp = down + (size_t)p * H2 * W2 + (size_t)gi * W2;
#pragma unroll
        for (int v = 0; v < 8; ++v) {
            int gj = j0 + v + 8 * half;
            if (gj < W2) dp[gj] = dacc[v];
        }
    }
}

// ---------------------------------------------------------------------------
// Fused: up-sample (2x dilated 5-tap, factored to 3x3 taps on down), z = x-up,
// 5x5 divisive-normalization pool on |z|, out = z / (pool + sigma).
// 256 threads compute a 32x32 output tile; z kept in LDS with a 2-px halo.
// ---------------------------------------------------------------------------
__global__ void __launch_bounds__(256)
nlp_sub(const float* __restrict__ src, const float* __restrict__ down,
        float* __restrict__ out, int H, int W, int gammaFlag) {
    __shared__ float zb[36][37];
    __shared__ float dt[20][21];

    const int H2 = H / 2 + 2, W2 = W / 2 + 2;
    const int p  = blockIdx.z;
    const int y0 = blockIdx.y * 32, x0 = blockIdx.x * 32;
    const int tid = threadIdx.x;
    const float* dw = down + (size_t)p * H2 * W2;

    const int dr0 = (y0 >> 1) - 1, dc0 = (x0 >> 1) - 1;
    __builtin_prefetch(dw + (size_t)(dr0 < 0 ? 0 : dr0) * W2, 0, 1);

    // stage the 20x20 down sub-tile feeding this 32x32 region
    for (int t = tid; t < 400; t += 256) {
        int ly = t / 20, lx = t % 20;
        int r = dr0 + ly; r = r < 0 ? 0 : (r >= H2 ? H2 - 1 : r);
        int c = dc0 + lx; c = c < 0 ? 0 : (c >= W2 ? W2 - 1 : c);
        dt[ly][lx] = dw[(size_t)r * W2 + c];
    }
    __syncthreads();

    // z over 36x36 (2-px halo); zero outside the image (== SAME zero-pad)
    for (int t = tid; t < 1296; t += 256) {
        int ly = t / 36, lx = t % 36;
        int y = y0 - 2 + ly, x = x0 - 2 + lx;
        float z = 0.0f;
        if (y >= 0 && y < H && x >= 0 && x < W) {
            float xv = src_at(src, p, y, x, H, W, gammaFlag);
            float wv[3], wh[3];
            int r0, c0;
            if (y & 1) { r0 = (y + 1) >> 1; wv[0] = 0.5f; wv[1] = 0.5f; wv[2] = 0.0f; }
            else       { r0 = y >> 1;       wv[0] = 0.1f; wv[1] = 0.8f; wv[2] = 0.1f; }
            if (x & 1) { c0 = (x + 1) >> 1; wh[0] = 0.5f; wh[1] = 0.5f; wh[2] = 0.0f; }
            else       { c0 = x >> 1;       wh[0] = 0.1f; wh[1] = 0.8f; wh[2] = 0.1f; }
            float up = 0.0f;
#pragma unroll
            for (int rr = 0; rr < 3; ++rr) {
                int lr = r0 + rr - dr0; lr = lr < 0 ? 0 : (lr > 19 ? 19 : lr);
                float rowv = 0.0f;
#pragma unroll
                for (int cc = 0; cc < 3; ++cc) {
                    int lc = c0 + cc - dc0; lc = lc < 0 ? 0 : (lc > 19 ? 19 : lc);
                    rowv = fmaf(wh[cc], dt[lr][lc], rowv);
                }
                up = fmaf(wv[rr], rowv, up);
            }
            z = xv - up;
        }
        zb[ly][lx] = z;
    }
    __syncthreads();

    const float Pk[5][5] = {
        {0.04f, 0.04f, 0.05f, 0.04f, 0.04f},
        {0.04f, 0.03f, 0.04f, 0.03f, 0.04f},
        {0.05f, 0.04f, 0.05f, 0.04f, 0.05f},
        {0.04f, 0.03f, 0.04f, 0.03f, 0.04f},
        {0.04f, 0.04f, 0.05f, 0.04f, 0.04f}};

    const int b = p / 3, ch = p - 3 * b;
#pragma unroll
    for (int q = 0; q < 4; ++q) {
        int idx = tid + q * 256;
        int oy = idx >> 5, ox = idx & 31;
        float zc = zb[oy + 2][ox + 2];
        float pool = 0.0f;
#pragma unroll
        for (int dy = 0; dy < 5; ++dy)
#pragma unroll
            for (int dx = 0; dx < 5; ++dx)
                pool = fmaf(Pk[dy][dx], fabsf(zb[oy + dy][ox + dx]), pool);
        float o = zc / (pool + SIGMA_P);
        int gy = y0 + oy, gx = x0 + ox;                       // exact tiling
        out[(((size_t)b * H + gy) * W + gx) * 3 + ch] = o;    // NHWC
    }
}

// lowpass residual: x5 = crop(down4); out = x5 / (|x5| + 4.86)
__global__ void __launch_bounds__(256)
nlp_resid(const float* __restrict__ d4, float* __restrict__ out) {
    const int p = blockIdx.x, tid = threadIdx.x;      // 256 = 16*16
    const int y = tid >> 4, x = tid & 15;
    float v = d4[(size_t)p * 18 * 18 + (size_t)(y + 1) * 18 + (x + 1)];
    float o = v / (fabsf(v) + SIGMA_LP);
    int b = p / 3, c = p - 3 * b;
    out[(((size_t)b * 16 + y) * 16 + x) * 3 + c] = o;
}

extern "C" void kernel_launch(void* const* d_in, const int* in_sizes, int n_in,
                              void* d_out, int out_size, void* d_ws, size_t ws_size,
                              hipStream_t stream) {
    (void)in_sizes; (void)n_in; (void)out_size; (void)ws_size;
    const float* img = (const float*)d_in[0];     // 16 x 512 x 512 x 3 fp32
    float* out = (float*)d_out;
    float* ws  = (float*)d_ws;

    const int Hl[6] = {512, 256, 128, 64, 32, 16};

    // workspace: 5 down buffers, plane dims (H/2+2)^2, 48 planes each (~17 MB)
    size_t dof[5]; size_t off = 0;
    for (int l = 0; l < 5; ++l) {
        dof[l] = off;
        size_t h2 = (size_t)(Hl[l] / 2 + 2);
        off += (size_t)48 * h2 * h2;
    }
    // output offsets: tuple concatenated flat in return order, NHWC each
    size_t oof[6]; size_t o2 = 0;
    for (int l = 0; l < 6; ++l) {
        oof[l] = o2;
        o2 += (size_t)16 * Hl[l] * Hl[l] * 3;
    }

    const float* srcp = img;
    int gam = 1;
    for (int l = 0; l < 5; ++l) {
        const int H = Hl[l];
        const int H2 = H / 2 + 2;
        const int T = (H2 + 15) / 16;
        nlp_down<<<dim3(T, T, 48), 32, 0, stream>>>(srcp, ws + dof[l], H, H, gam);
        nlp_sub<<<dim3(H / 32, H / 32, 48), 256, 0, stream>>>(
            srcp, ws + dof[l], out + oof[l], H, H, gam);
        srcp = ws + dof[l];
        gam = 0;
    }
    nlp_resid<<<48, 256, 0, stream>>>(ws + dof[4], out + oof[5]);
}